// RGCNBasisLayer_4629974745758
// MI455X (gfx1250) — compile-verified
//
#include <hip/hip_runtime.h>

typedef __attribute__((ext_vector_type(2))) float v2f;
typedef __attribute__((ext_vector_type(8))) float v8f;

#define N_NODES   50000
#define N_EDGES   800000
#define FEAT_DIM  32
#define NODE_TILES (N_NODES / 16)   // 3125
#define EDGE_TILES (N_EDGES / 16)   // 50000
#define WAVES_PER_BLOCK 8

// D = A(16x4 f32) * B(4x16 f32) + C(16x16 f32)
#define WMMA_F32(acc, a, b) \
  acc = __builtin_amdgcn_wmma_f32_16x16x4_f32(false, (a), false, (b), (short)0, (acc), false, false)

// Pack a [4K x 32] row-major B matrix into fragment order:
//   packed[(k*2 + h)*32 + n] = { B[4k+2h][n], B[4k+2h+1][n] }
// so each wave lane's SRC1 fragment is ONE aligned ds_load_b64.
__device__ __forceinline__ void stage_packed(v2f* dst, const float* src, int ksteps) {
  const int total = ksteps * 64;
  for (int e = threadIdx.x; e < total; e += blockDim.x) {
    const int n  = e & 31;
    const int h  = (e >> 5) & 1;
    const int kk = e >> 6;
    const int r  = 4 * kk + 2 * h;
    v2f t;
    t.x = src[r * 32 + n];
    t.y = src[(r + 1) * 32 + n];
    dst[e] = t;
  }
}

__global__ void zero_kernel(float* __restrict__ p, int n) {
  int i = blockIdx.x * blockDim.x + threadIdx.x;
  if (i < n) p[i] = 0.0f;
}

// init_fea = concat(feat, embed[idx]) @ transform   [N,32], also writes out[:,0,:]
__global__ void init_fea_kernel(const float* __restrict__ feat,
                                const float* __restrict__ embed,
                                const float* __restrict__ transform,   // [64][32]
                                const int*   __restrict__ idx,
                                float* __restrict__ init_fea,
                                float* __restrict__ out) {
  __shared__ v2f sT[16 * 64];              // 16 K-steps, fragment-packed
  stage_packed(sT, transform, 16);
  __syncthreads();

  const int lane = threadIdx.x & 31;
  const int wave = threadIdx.x >> 5;
  const int tile = blockIdx.x * WAVES_PER_BLOCK + wave;
  if (tile >= NODE_TILES) return;          // wave-uniform exit
  const int row  = lane & 15;
  const int half = lane >> 4;
  const int col  = row;

  const int node = tile * 16 + row;
  const float* frow = feat  + (size_t)node * FEAT_DIM;
  const float* erow = embed + (size_t)idx[node] * FEAT_DIM;

  v2f fa[16];
#pragma unroll
  for (int k = 0; k < 8; ++k) fa[k]     = ((const v2f*)frow)[2 * k + half];
#pragma unroll
  for (int k = 0; k < 8; ++k) fa[8 + k] = ((const v2f*)erow)[2 * k + half];

  v8f acc0 = {}, acc1 = {};
#pragma unroll
  for (int k = 0; k < 16; ++k) {
    const v2f* pB = sT + (k * 2 + half) * 32;
    WMMA_F32(acc0, fa[k], pB[col]);
    WMMA_F32(acc1, fa[k], pB[col + 16]);
  }

  const int mbase = tile * 16 + 8 * half;
#pragma unroll
  for (int j = 0; j < 8; ++j) {
    const int m = mbase + j;
    init_fea[(size_t)m * 32 + col]      = acc0[j];
    init_fea[(size_t)m * 32 + col + 16] = acc1[j];
    out[(size_t)m * 64 + col]      = acc0[j];   // out[:,0,:]
    out[(size_t)m * 64 + col + 16] = acc1[j];
  }
}

// Per-edge: msg + attention gate + atomic scatter into agg. 16 edges per wave.
__global__ void edge_kernel(const float* __restrict__ init_fea,
                            const float* __restrict__ weight,    // [4][32][32]
                            const float* __restrict__ w_comp,    // [16][4]
                            const float* __restrict__ A_w,       // [96][32]
                            const float* __restrict__ A_b,       // [32]
                            const float* __restrict__ B_w,       // [32]
                            const float* __restrict__ B_b,       // [1]
                            const float* __restrict__ attn_emb,  // [16][32]
                            const int*   __restrict__ edge_src,
                            const int*   __restrict__ edge_dst,
                            const int*   __restrict__ edge_type,
                            float* __restrict__ agg) {
  __shared__ v2f   sW[4 * 8 * 64];   // basis weights, fragment-packed (16 KB)
  __shared__ v2f   sA[24 * 64];      // A_w, fragment-packed (12 KB)
  __shared__ v2f   sE[16 * 16];      // attn_emb as v2f pairs (2 KB)
  __shared__ float sC[16 * 4];       // w_comp
#pragma unroll
  for (int b = 0; b < 4; ++b) stage_packed(sW + b * 512, weight + b * 1024, 8);
  stage_packed(sA, A_w, 24);
  for (int i = threadIdx.x; i < 16 * 16; i += blockDim.x) sE[i] = ((const v2f*)attn_emb)[i];
  for (int i = threadIdx.x; i < 64;      i += blockDim.x) sC[i] = w_comp[i];
  __syncthreads();

  const int lane = threadIdx.x & 31;
  const int wave = threadIdx.x >> 5;
  const int tile = blockIdx.x * WAVES_PER_BLOCK + wave;
  if (tile >= EDGE_TILES) return;          // wave-uniform exit
  const int base = tile * 16;
  const int row  = lane & 15;
  const int half = lane >> 4;
  const int col  = row;

  // Per-row gathers for A fragments (rows of the 16-edge tile)
  const int src_r = edge_src[base + row];
  const int dst_r = edge_dst[base + row];
  const int typ_r = edge_type[base + row];
  const float* xrow = init_fea + (size_t)src_r * 32;
  const float* yrow = init_fea + (size_t)dst_r * 32;

  v2f fx[8], fy[8], fe[8];
#pragma unroll
  for (int k = 0; k < 8; ++k) {
    fx[k] = ((const v2f*)xrow)[2 * k + half];
    fy[k] = ((const v2f*)yrow)[2 * k + half];
    fe[k] = sE[typ_r * 16 + 2 * k + half];
  }

  // Per-j (D-layout rows this lane owns: M = j + 8*half) edge metadata
  int dstj[8], tj[8];
#pragma unroll
  for (int j = 0; j < 8; ++j) {
    dstj[j] = edge_dst[base + j + 8 * half];
    tj[j]   = edge_type[base + j + 8 * half];
  }

  // ---- attention MLP: z = relu([x,y,emb] @ A_w + A_b), K=96 ----
  const float ab0 = A_b[col], ab1 = A_b[col + 16];
  v8f z0, z1;
#pragma unroll
  for (int j = 0; j < 8; ++j) { z0[j] = ab0; z1[j] = ab1; }
#pragma unroll
  for (int k = 0; k < 24; ++k) {
    v2f a = (k < 8) ? fx[k] : ((k < 16) ? fy[k - 8] : fe[k - 16]);
    const v2f* pB = sA + (k * 2 + half) * 32;
    WMMA_F32(z0, a, pB[col]);
    WMMA_F32(z1, a, pB[col + 16]);
  }

  // gate: a = sigmoid(relu(z) . B_w + B_b); butterfly-reduce over N (16 lanes/half)
  const float bw0 = B_w[col], bw1 = B_w[col + 16];
  const float bb  = B_b[0];
  float aj[8];
#pragma unroll
  for (int j = 0; j < 8; ++j) {
    float p = fmaxf(z0[j], 0.0f) * bw0 + fmaxf(z1[j], 0.0f) * bw1;
    p += __shfl_xor(p, 1);
    p += __shfl_xor(p, 2);
    p += __shfl_xor(p, 4);
    p += __shfl_xor(p, 8);
    aj[j] = 1.0f / (1.0f + __expf(-(p + bb)));
  }

  // ---- msg = sum_b w_comp[t,b] * (X @ weight[b]) ----
  v8f m0 = {}, m1 = {};
#pragma unroll
  for (int b = 0; b < 4; ++b) {
    v8f t0 = {}, t1 = {};
    const v2f* Wb = sW + b * 512;
#pragma unroll
    for (int k = 0; k < 8; ++k) {
      const v2f* pB = Wb + (k * 2 + half) * 32;
      WMMA_F32(t0, fx[k], pB[col]);
      WMMA_F32(t1, fx[k], pB[col + 16]);
    }
#pragma unroll
    for (int j = 0; j < 8; ++j) {
      const float wc = sC[tj[j] * 4 + b];
      m0[j] += wc * t0[j];
      m1[j] += wc * t1[j];
    }
  }

  // scatter-reduce: agg[dst] += a * msg
#pragma unroll
  for (int j = 0; j < 8; ++j) {
    atomicAdd(&agg[(size_t)dstj[j] * 32 + col],      aj[j] * m0[j]);
    atomicAdd(&agg[(size_t)dstj[j] * 32 + col + 16], aj[j] * m1[j]);
  }
}

// h = relu(init_fea @ self_loop + agg) -> out[:,1,:]
__global__ void final_kernel(const float* __restrict__ init_fea,
                             const float* __restrict__ self_loop,  // [32][32]
                             const float* __restrict__ agg,
                             float* __restrict__ out) {
  __shared__ v2f sS[8 * 64];               // self_loop, fragment-packed
  stage_packed(sS, self_loop, 8);
  __syncthreads();

  const int lane = threadIdx.x & 31;
  const int wave = threadIdx.x >> 5;
  const int tile = blockIdx.x * WAVES_PER_BLOCK + wave;
  if (tile >= NODE_TILES) return;
  const int row  = lane & 15;
  const int half = lane >> 4;
  const int col  = row;

  const float* xrow = init_fea + (size_t)(tile * 16 + row) * 32;
  v2f fa[8];
#pragma unroll
  for (int k = 0; k < 8; ++k) fa[k] = ((const v2f*)xrow)[2 * k + half];

  v8f acc0 = {}, acc1 = {};
#pragma unroll
  for (int k = 0; k < 8; ++k) {
    const v2f* pB = sS + (k * 2 + half) * 32;
    WMMA_F32(acc0, fa[k], pB[col]);
    WMMA_F32(acc1, fa[k], pB[col + 16]);
  }

  const int mbase = tile * 16 + 8 * half;
#pragma unroll
  for (int j = 0; j < 8; ++j) {
    const int m = mbase + j;
    const float h0 = fmaxf(acc0[j] + agg[(size_t)m * 32 + col],      0.0f);
    const float h1 = fmaxf(acc1[j] + agg[(size_t)m * 32 + col + 16], 0.0f);
    out[(size_t)m * 64 + 32 + col]      = h0;    // out[:,1,:]
    out[(size_t)m * 64 + 32 + col + 16] = h1;
  }
}

extern "C" void kernel_launch(void* const* d_in, const int* in_sizes, int n_in,
                              void* d_out, int out_size, void* d_ws, size_t ws_size,
                              hipStream_t stream) {
  const float* feat      = (const float*)d_in[0];
  const float* embed     = (const float*)d_in[1];
  const float* transform = (const float*)d_in[2];
  const float* weight    = (const float*)d_in[3];
  const float* w_comp    = (const float*)d_in[4];
  const float* self_loop = (const float*)d_in[5];
  const float* A_w       = (const float*)d_in[6];
  const float* A_b       = (const float*)d_in[7];
  const float* B_w       = (const float*)d_in[8];
  const float* B_b       = (const float*)d_in[9];
  const float* attn_emb  = (const float*)d_in[10];
  const int*   idx       = (const int*)d_in[11];
  const int*   esrc      = (const int*)d_in[12];
  const int*   edst      = (const int*)d_in[13];
  const int*   etyp      = (const int*)d_in[14];
  float* out = (float*)d_out;

  float* init_fea = (float*)d_ws;                       // N*32 floats
  float* agg      = init_fea + (size_t)N_NODES * 32;    // N*32 floats

  const int aggN = N_NODES * 32;
  zero_kernel<<<(aggN + 255) / 256, 256, 0, stream>>>(agg, aggN);

  const int nodeBlocks = (NODE_TILES + WAVES_PER_BLOCK - 1) / WAVES_PER_BLOCK;  // 391
  init_fea_kernel<<<nodeBlocks, 32 * WAVES_PER_BLOCK, 0, stream>>>(
      feat, embed, transform, idx, init_fea, out);

  const int edgeBlocks = (EDGE_TILES + WAVES_PER_BLOCK - 1) / WAVES_PER_BLOCK;  // 6250
  edge_kernel<<<edgeBlocks, 32 * WAVES_PER_BLOCK, 0, stream>>>(
      init_fea, weight, w_comp, A_w, A_b, B_w, B_b, attn_emb,
      esrc, edst, etyp, agg);

  final_kernel<<<nodeBlocks, 32 * WAVES_PER_BLOCK, 0, stream>>>(
      init_fea, self_loop, agg, out);
}